// GCNBoard_43164421325166
// MI455X (gfx1250) — compile-verified
//
#include <hip/hip_runtime.h>
#include <hip/hip_bf16.h>
#include <stdint.h>

typedef float v2f __attribute__((ext_vector_type(2)));
typedef float v8f __attribute__((ext_vector_type(8)));

// ---------------- degree / normalization ----------------
__global__ void k_deg_init(float* deg, int n) {
  int i = blockIdx.x * blockDim.x + threadIdx.x;
  if (i < n) deg[i] = 1.0f;  // self-loop contributes 1
}
__global__ void k_deg_accum(const long long* __restrict__ row, float* deg, int ne) {
  int i = blockIdx.x * blockDim.x + threadIdx.x;
  if (i < ne) atomicAdd(&deg[(int)row[i]], 1.0f);
}
__global__ void k_deg_rsqrt(float* deg, int n) {
  int i = blockIdx.x * blockDim.x + threadIdx.x;
  if (i < n) deg[i] = rsqrtf(deg[i]);
}
__global__ void k_zero_g(unsigned* g) { g[threadIdx.x] = 0u; }

// ---------------- WMMA fp32 GEMM:  out[N,256] = A[N,256] @ W[256,256] + b ----------------
// MODE 0: A = concat(x[N,128], feat[N,128]); MODE 1: A = relu(H[N,256])
template <int MODE>
__global__ __launch_bounds__(256) void k_lin_wmma(
    const float* __restrict__ A0, const float* __restrict__ A1,
    const float* __restrict__ W, const float* __restrict__ bias,
    float* __restrict__ out, int nrows)
{
  __shared__ float Wl[64 * 256];  // 64 KB: K-chunk of W staged in LDS
  const int t    = threadIdx.x;
  const int wave = t >> 5;
  const int lane = t & 31;
  const int half = lane >> 4;   // 0: K={0,1}, 1: K={2,3} of the 16x4 A frag
  const int l16  = lane & 15;
  const int mbase = blockIdx.x * 128 + wave * 16;
  const int mrow  = mbase + l16;

  v8f acc[16];
  v8f z = {0.f,0.f,0.f,0.f,0.f,0.f,0.f,0.f};
#pragma unroll
  for (int i = 0; i < 16; ++i) acc[i] = z;

  for (int kb = 0; kb < 4; ++kb) {
    // cooperative LDS stage of W rows [kb*64, kb*64+64)
    const float4* Wg = (const float4*)(W + (size_t)kb * 64 * 256);
    float4* Wl4 = (float4*)Wl;
#pragma unroll
    for (int i = 0; i < 16; ++i) Wl4[i * 256 + t] = Wg[i * 256 + t];
    __syncthreads();

#pragma unroll
    for (int s = 0; s < 16; ++s) {
      const int k4 = kb * 64 + s * 4;
      const int k0 = k4 + half * 2;
      v2f a;
      if (mrow < nrows) {
        if (MODE == 0) {
          const float* p = (k0 < 128) ? (A0 + (size_t)mrow * 128 + k0)
                                      : (A1 + (size_t)mrow * 128 + (k0 - 128));
          a.x = p[0]; a.y = p[1];
        } else {
          const float* p = A0 + (size_t)mrow * 256 + k0;
          a.x = fmaxf(p[0], 0.f); a.y = fmaxf(p[1], 0.f);  // fused relu
        }
      } else { a.x = 0.f; a.y = 0.f; }

      const int kk = s * 4 + half * 2;
#pragma unroll
      for (int nt = 0; nt < 16; ++nt) {
        v2f b;
        b.x = Wl[kk * 256 + nt * 16 + l16];
        b.y = Wl[(kk + 1) * 256 + nt * 16 + l16];
        acc[nt] = __builtin_amdgcn_wmma_f32_16x16x4_f32(
            false, a, false, b, (short)0, acc[nt], false, false);
      }
    }
    __syncthreads();
  }

#pragma unroll
  for (int nt = 0; nt < 16; ++nt) {
    const int n = nt * 16 + l16;
    const float bn = bias[n];
#pragma unroll
    for (int r = 0; r < 8; ++r) {
      const int m = mbase + half * 8 + r;
      if (m < nrows) out[(size_t)m * 256 + n] = acc[nt][r] + bn;
    }
  }
}

// ---------------- self-loop term doubles as zero-init:  dst = dis[i]^2 * src ----------------
__global__ void k_selfloop(const float* __restrict__ dis, const float* __restrict__ src,
                           float* __restrict__ dst, int n) {
  int idx = blockIdx.x * blockDim.x + threadIdx.x;  // one float4 each
  if (idx < n * 64) {
    int node = idx >> 6;
    float d = dis[node];
    float w = d * d;
    float4 v = ((const float4*)src)[idx];
    float4 o; o.x = w*v.x; o.y = w*v.y; o.z = w*v.z; o.w = w*v.w;
    ((float4*)dst)[idx] = o;
  }
}

// ---------------- edge scatter: one wave per edge, dst[col] += norm * src[row] ----------------
__global__ __launch_bounds__(256) void k_scatter(
    const long long* __restrict__ row, const long long* __restrict__ col,
    const float* __restrict__ dis, const float* __restrict__ src,
    float* __restrict__ dst, int ne)
{
  int w = (int)((blockIdx.x * blockDim.x + threadIdx.x) >> 5);
  int lane = threadIdx.x & 31;
  if (w >= ne) return;
  int r = (int)row[w], c = (int)col[w];
  float norm = dis[r] * dis[c];
  const float4* s4 = (const float4*)(src + (size_t)r * 256);
  float* d = dst + (size_t)c * 256;
#pragma unroll
  for (int i = 0; i < 2; ++i) {
    int f4 = lane + i * 32;
    float4 v = s4[f4];
    int f = f4 * 4;
    atomicAdd(d + f + 0, norm * v.x);
    atomicAdd(d + f + 1, norm * v.y);
    atomicAdd(d + f + 2, norm * v.z);
    atomicAdd(d + f + 3, norm * v.w);
  }
}

// ---------------- global max pool over relu(h); relu>=0 so u32 bits order-preserving ----------------
__global__ void k_maxpool(const float* __restrict__ h, unsigned* __restrict__ g, int n) {
  int f = threadIdx.x;  // 256 features
  float m = 0.f;
  for (int r = blockIdx.x; r < n; r += gridDim.x)
    m = fmaxf(m, h[(size_t)r * 256 + f]);
  atomicMax(g + f, __float_as_uint(m));
}

// ---------------- heads: finx = g@Wp+bp, softmax, finy = g@Wq+bq ----------------
__global__ void k_heads(const float* __restrict__ g, const float* __restrict__ Wp,
                        const float* __restrict__ bp, const float* __restrict__ Wq,
                        const float* __restrict__ bq, float* __restrict__ out) {
  __shared__ float gs[256];
  __shared__ float red[128];
  int t = threadIdx.x;  // 128 threads
  gs[t] = g[t]; gs[t + 128] = g[t + 128];
  __syncthreads();
  float fx = -3.4e38f;
  if (t < 101) {
    float s = bp[t];
    for (int k = 0; k < 256; ++k) s += gs[k] * Wp[k * 101 + t];
    out[t] = s; fx = s;
  }
  red[t] = fx; __syncthreads();
  for (int off = 64; off > 0; off >>= 1) {
    if (t < off) red[t] = fmaxf(red[t], red[t + off]);
    __syncthreads();
  }
  float mx = red[0]; __syncthreads();
  float e = (t < 101) ? __expf(fx - mx) : 0.f;
  red[t] = e; __syncthreads();
  for (int off = 64; off > 0; off >>= 1) {
    if (t < off) red[t] += red[t + off];
    __syncthreads();
  }
  float inv = 1.f / red[0];
  if (t < 101) out[101 + t] = e * inv;
  if (t == 0) {
    float s = bq[0];
    for (int k = 0; k < 256; ++k) s += gs[k] * Wq[k];
    out[202] = s;
  }
}

extern "C" void kernel_launch(void* const* d_in, const int* in_sizes, int n_in,
                              void* d_out, int out_size, void* d_ws, size_t ws_size,
                              hipStream_t stream) {
  const float*     x    = (const float*)d_in[0];
  const float*     feat = (const float*)d_in[1];
  const long long* ei   = (const long long*)d_in[2];
  const float* W1 = (const float*)d_in[3];
  const float* b1 = (const float*)d_in[4];
  const float* W2 = (const float*)d_in[5];
  const float* b2 = (const float*)d_in[6];
  const float* Wp = (const float*)d_in[7];
  const float* bp = (const float*)d_in[8];
  const float* Wq = (const float*)d_in[9];
  const float* bq = (const float*)d_in[10];

  const int N = in_sizes[0] / 128;   // 50000
  const int E = in_sizes[2] / 2;     // 800000
  const long long* row = ei;
  const long long* col = ei + E;

  float* ws  = (float*)d_ws;
  const size_t npad = ((size_t)N + 127) / 128 * 128;
  float* dis = ws;                           // [N]
  float* B0  = ws + npad;                    // [N*256] linear output
  float* B1  = B0 + (size_t)N * 256;         // [N*256] aggregation
  unsigned* g = (unsigned*)(B1 + (size_t)N * 256);  // [256]

  const int T = 256;
  // degree / dis
  k_deg_init  <<<(N + T - 1) / T, T, 0, stream>>>(dis, N);
  k_deg_accum <<<(E + T - 1) / T, T, 0, stream>>>(row, dis, E);
  k_deg_rsqrt <<<(N + T - 1) / T, T, 0, stream>>>(dis, N);
  k_zero_g    <<<1, 256, 0, stream>>>(g);

  const int gemm_blocks = (N + 127) / 128;
  const int sl_blocks   = (N * 64 + T - 1) / T;
  const int sc_blocks   = (E + 7) / 8;   // 8 waves/block, 1 edge/wave

  // layer 1
  k_lin_wmma<0><<<gemm_blocks, 256, 0, stream>>>(x, feat, W1, b1, B0, N);
  k_selfloop   <<<sl_blocks, T, 0, stream>>>(dis, B0, B1, N);
  k_scatter    <<<sc_blocks, T, 0, stream>>>(row, col, dis, B0, B1, E);
  // layer 2 (relu fused into A-load)
  k_lin_wmma<1><<<gemm_blocks, 256, 0, stream>>>(B1, nullptr, W2, b2, B0, N);
  k_selfloop   <<<sl_blocks, T, 0, stream>>>(dis, B0, B1, N);
  k_scatter    <<<sc_blocks, T, 0, stream>>>(row, col, dis, B0, B1, E);
  // pool + heads
  k_maxpool <<<1024, 256, 0, stream>>>(B1, g, N);
  k_heads   <<<1, 128, 0, stream>>>((const float*)g, Wp, bp, Wq, bq, (float*)d_out);
}